// SingleDetEncoder_11802570130429
// MI455X (gfx1250) — compile-verified
//
#include <hip/hip_runtime.h>
#include <hip/hip_bf16.h>

// ---------------------------------------------------------------------------
// HEALPix graph-CNN encoder (3x [ChebConv->ELU->ChebConv->ELU->avgpool4])
// GEMMs via V_WMMA_F32_16X16X4_F32, templated on (Cin, Cout).
// Two execution plans selected by ws_size:
//  - FUSED: materialize all K Chebyshev bases, one register-resident GEMM per
//    conv (bias + 10*Cin/4 chained WMMAs, single store) -> no per-k C-tile
//    round trips through HBM.
//  - STEPWISE: 3 rotating basis buffers, accumulate out per k (smaller ws).
// SpMV via coalesced gather + global float atomics, templated on C.
// ---------------------------------------------------------------------------

typedef __attribute__((ext_vector_type(2))) float v2f;
typedef __attribute__((ext_vector_type(8))) float v8f;

#define K_CHEB 10
#define BATCH  8

// ------------------------------ elementwise --------------------------------

__global__ void zero_kernel(float* __restrict__ y, long n) {
  long i = (long)blockIdx.x * blockDim.x + threadIdx.x;
  long s = (long)gridDim.x * blockDim.x;
  for (; i < n; i += s) y[i] = 0.0f;
}

__global__ void copy_kernel(float* __restrict__ y, const float* __restrict__ x, long n) {
  long i = (long)blockIdx.x * blockDim.x + threadIdx.x;
  long s = (long)gridDim.x * blockDim.x;
  for (; i < n; i += s) y[i] = x[i];
}

__global__ void negcopy_kernel(float* __restrict__ y, const float* __restrict__ x, long n) {
  long i = (long)blockIdx.x * blockDim.x + threadIdx.x;
  long s = (long)gridDim.x * blockDim.x;
  for (; i < n; i += s) y[i] = -x[i];
}

// Cout is a power of two -> mask instead of modulo.
__global__ void bias_init_kernel(float* __restrict__ out, const float* __restrict__ bias,
                                 long n, int coutMask) {
  long i = (long)blockIdx.x * blockDim.x + threadIdx.x;
  long s = (long)gridDim.x * blockDim.x;
  for (; i < n; i += s) out[i] = bias[(int)i & coutMask];
}

__global__ void elu_kernel(float* __restrict__ y, long n) {
  long i = (long)blockIdx.x * blockDim.x + threadIdx.x;
  long s = (long)gridDim.x * blockDim.x;
  for (; i < n; i += s) {
    float v = y[i];
    y[i] = (v > 0.0f) ? v : expm1f(v);
  }
}

// ------------------------------- SpMV --------------------------------------
// y[(b*V + rows[e])*C + c] += scale * vals[e] * x[(b*V + cols[e])*C + c]
// Thread index i = (e*BATCH + b)*C + c -> consecutive lanes share an edge and
// walk consecutive channels: coalesced gather and coalesced atomics.

template <int C>
__global__ void spmv_acc_kernel(float* __restrict__ y, const float* __restrict__ x,
                                const int* __restrict__ rows, const int* __restrict__ cols,
                                const float* __restrict__ vals, float scale,
                                int V, int E) {
  constexpr int CSHIFT = (C == 1) ? 0 : (C == 32) ? 5 : (C == 64) ? 6 : 7;
  long total = (long)E * BATCH * C;
  long i = (long)blockIdx.x * blockDim.x + threadIdx.x;
  long s = (long)gridDim.x * blockDim.x;
  for (; i < total; i += s) {
    int c = (int)i & (C - 1);
    long t = i >> CSHIFT;
    int b = (int)t & (BATCH - 1);
    int e = (int)(t >> 3);
    float w  = vals[e] * scale;
    float xv = x[(((long)b * V + cols[e]) << CSHIFT) + c];
    atomicAdd(&y[(((long)b * V + rows[e]) << CSHIFT) + c], w * xv);
  }
}

// ------------------------------ pooling ------------------------------------
// in [B, 4*Vq, C] -> out [B, Vq, C], mean over 4 consecutive vertices.

template <int C>
__global__ void pool4_kernel(float* __restrict__ out, const float* __restrict__ in, int Vq) {
  constexpr int CSHIFT = (C == 32) ? 5 : (C == 64) ? 6 : 7;
  long total = (long)BATCH * Vq * C;
  long i = (long)blockIdx.x * blockDim.x + threadIdx.x;
  long s = (long)gridDim.x * blockDim.x;
  for (; i < total; i += s) {
    int c = (int)i & (C - 1);
    long t = i >> CSHIFT;
    int v = (int)(t % Vq);
    int b = (int)(t / Vq);
    const float* p = in + ((((long)b * Vq + v) * 4) << CSHIFT) + c;
    out[i] = 0.25f * (p[0] + p[C] + p[2 * C] + p[3 * C]);
  }
}

// --------------------------- WMMA GEMM (fp32) ------------------------------
// V_WMMA_F32_16X16X4_F32 layouts (ISA 7.12.2):
//   A 16x4 : lane&15 = M row; VGPR0/1 hold K = 2*(lane>>4) + {0,1}
//   B 4x16 : lane&15 = N col; VGPR0/1 hold K = 2*(lane>>4) + {0,1}
//   C 16x16: lane&15 = N col; VGPR g holds M = g + 8*(lane>>4)

// Per-k accumulate: out[R,COUT] += X[R,CIN] @ W[CIN,COUT]
template <int CIN, int COUT>
__global__ void wmma_gemm_acc_kernel(float* __restrict__ out, const float* __restrict__ X,
                                     const float* __restrict__ W, int Rtiles) {
  constexpr int NTILE = COUT >> 4;
  int lane = threadIdx.x & 31;
  int wave = threadIdx.x >> 5;
  int T = blockIdx.x * (blockDim.x >> 5) + wave;   // wave-uniform
  if (T >= Rtiles * NTILE) return;                 // whole wave exits together

  int tN = T & (NTILE - 1);
  int tM = T / NTILE;
  int n    = lane & 15;
  int kk   = (lane >> 4) << 1;
  int mOff = (lane >> 4) << 3;
  long rowBase = (long)tM * 16;
  int  colBase = tN * 16;

  float* outTile = out + (rowBase + mOff) * COUT + colBase + n;
  const float* wCol = W + colBase + n;

  v8f c;
#pragma unroll
  for (int g = 0; g < 8; ++g)
    c[g] = outTile[g * COUT];

  if constexpr (CIN == 1) {
    v2f a, b;
    a.x = (kk == 0) ? X[rowBase + n] : 0.0f;
    a.y = 0.0f;
    b.x = (kk == 0) ? wCol[0] : 0.0f;
    b.y = 0.0f;
    c = __builtin_amdgcn_wmma_f32_16x16x4_f32(false, a, false, b, (short)0, c,
                                              false, false);
  } else {
    const float* xrow = X + (rowBase + n) * CIN + kk;
#pragma unroll
    for (int ci = 0; ci < CIN; ci += 4) {
      v2f a = *(const v2f*)(xrow + ci);
      v2f b;
      b.x = wCol[(ci + kk) * COUT];
      b.y = wCol[(ci + kk + 1) * COUT];
      c = __builtin_amdgcn_wmma_f32_16x16x4_f32(false, a, false, b, (short)0, c,
                                                false, false);
    }
  }

#pragma unroll
  for (int g = 0; g < 8; ++g)
    outTile[g * COUT] = c[g];
}

// Fused over all K bases: out = bias + sum_k bases[k] @ W[k]; one store.
template <int CIN, int COUT>
__global__ void wmma_gemm_fused_kernel(float* __restrict__ out,
                                       const float* __restrict__ bases,  // [K][R][CIN]
                                       long baseStride,                  // R*CIN
                                       const float* __restrict__ W,      // [K][CIN][COUT]
                                       const float* __restrict__ bias,
                                       int Rtiles) {
  constexpr int NTILE = COUT >> 4;
  int lane = threadIdx.x & 31;
  int wave = threadIdx.x >> 5;
  int T = blockIdx.x * (blockDim.x >> 5) + wave;   // wave-uniform
  if (T >= Rtiles * NTILE) return;                 // whole wave exits together

  int tN = T & (NTILE - 1);
  int tM = T / NTILE;
  int n    = lane & 15;
  int kk   = (lane >> 4) << 1;
  int mOff = (lane >> 4) << 3;
  long rowBase = (long)tM * 16;
  int  colBase = tN * 16;

  v8f c;
  float bv = bias[colBase + n];
#pragma unroll
  for (int g = 0; g < 8; ++g)
    c[g] = bv;

  const float* wCol = W + colBase + n;
#pragma unroll 1
  for (int k = 0; k < K_CHEB; ++k) {
    if constexpr (CIN == 1) {
      v2f a, b;
      a.x = (kk == 0) ? bases[k * baseStride + rowBase + n] : 0.0f;
      a.y = 0.0f;
      b.x = (kk == 0) ? wCol[k * CIN * COUT] : 0.0f;
      b.y = 0.0f;
      c = __builtin_amdgcn_wmma_f32_16x16x4_f32(false, a, false, b, (short)0, c,
                                                false, false);
    } else {
      const float* xrow = bases + k * baseStride + (rowBase + n) * CIN + kk;
      const float* wk = wCol + k * CIN * COUT;
#pragma unroll
      for (int ci = 0; ci < CIN; ci += 4) {
        v2f a = *(const v2f*)(xrow + ci);
        v2f b;
        b.x = wk[(ci + kk) * COUT];
        b.y = wk[(ci + kk + 1) * COUT];
        c = __builtin_amdgcn_wmma_f32_16x16x4_f32(false, a, false, b, (short)0, c,
                                                  false, false);
      }
    }
  }

  float* outTile = out + (rowBase + mOff) * COUT + colBase + n;
#pragma unroll
  for (int g = 0; g < 8; ++g)
    outTile[g * COUT] = c[g];
}

// ------------------------------ host side ----------------------------------

static inline int nblk(long n, int t) {
  long b = (n + t - 1) / t;
  if (b > (1L << 20)) b = (1L << 20);
  if (b < 1) b = 1;
  return (int)b;
}

static void launch_gemm_acc(float* out, const float* X, const float* W,
                            int Rtiles, int Cin, int Cout, hipStream_t stream) {
  int blocks = (Rtiles * (Cout >> 4) + 7) / 8;
  if (Cin == 1 && Cout == 32)
    wmma_gemm_acc_kernel<1, 32><<<blocks, 256, 0, stream>>>(out, X, W, Rtiles);
  else if (Cin == 32 && Cout == 32)
    wmma_gemm_acc_kernel<32, 32><<<blocks, 256, 0, stream>>>(out, X, W, Rtiles);
  else if (Cin == 32 && Cout == 64)
    wmma_gemm_acc_kernel<32, 64><<<blocks, 256, 0, stream>>>(out, X, W, Rtiles);
  else if (Cin == 64 && Cout == 64)
    wmma_gemm_acc_kernel<64, 64><<<blocks, 256, 0, stream>>>(out, X, W, Rtiles);
  else if (Cin == 64 && Cout == 128)
    wmma_gemm_acc_kernel<64, 128><<<blocks, 256, 0, stream>>>(out, X, W, Rtiles);
  else if (Cin == 128 && Cout == 128)
    wmma_gemm_acc_kernel<128, 128><<<blocks, 256, 0, stream>>>(out, X, W, Rtiles);
}

static void launch_gemm_fused(float* out, const float* bases, long baseStride,
                              const float* W, const float* bias,
                              int Rtiles, int Cin, int Cout, hipStream_t stream) {
  int blocks = (Rtiles * (Cout >> 4) + 7) / 8;
  if (Cin == 1 && Cout == 32)
    wmma_gemm_fused_kernel<1, 32><<<blocks, 256, 0, stream>>>(out, bases, baseStride, W, bias, Rtiles);
  else if (Cin == 32 && Cout == 32)
    wmma_gemm_fused_kernel<32, 32><<<blocks, 256, 0, stream>>>(out, bases, baseStride, W, bias, Rtiles);
  else if (Cin == 32 && Cout == 64)
    wmma_gemm_fused_kernel<32, 64><<<blocks, 256, 0, stream>>>(out, bases, baseStride, W, bias, Rtiles);
  else if (Cin == 64 && Cout == 64)
    wmma_gemm_fused_kernel<64, 64><<<blocks, 256, 0, stream>>>(out, bases, baseStride, W, bias, Rtiles);
  else if (Cin == 64 && Cout == 128)
    wmma_gemm_fused_kernel<64, 128><<<blocks, 256, 0, stream>>>(out, bases, baseStride, W, bias, Rtiles);
  else if (Cin == 128 && Cout == 128)
    wmma_gemm_fused_kernel<128, 128><<<blocks, 256, 0, stream>>>(out, bases, baseStride, W, bias, Rtiles);
}

static void launch_spmv(float* y, const float* x, const int* rows, const int* cols,
                        const float* vals, float scale, int V, int Cin, int E,
                        hipStream_t stream) {
  long spN = (long)E * BATCH * Cin;
  int blocks = nblk(spN, 256);
  if (Cin == 1)
    spmv_acc_kernel<1><<<blocks, 256, 0, stream>>>(y, x, rows, cols, vals, scale, V, E);
  else if (Cin == 32)
    spmv_acc_kernel<32><<<blocks, 256, 0, stream>>>(y, x, rows, cols, vals, scale, V, E);
  else if (Cin == 64)
    spmv_acc_kernel<64><<<blocks, 256, 0, stream>>>(y, x, rows, cols, vals, scale, V, E);
  else if (Cin == 128)
    spmv_acc_kernel<128><<<blocks, 256, 0, stream>>>(y, x, rows, cols, vals, scale, V, E);
}

// FUSED plan: all K bases materialized in `bases`, then one GEMM.
static void run_cheb_conv_fused(const float* Xin, const float* W, const float* bias,
                                const int* rows, const int* cols, const float* vals,
                                float* out, float* bases,
                                int V, int E, int Cin, int Cout, hipStream_t stream) {
  long R = (long)V * BATCH;
  int Rtiles = (int)(R >> 4);
  long S = R * Cin;   // stride between bases

  // B0 = x ; B1 = L B0
  copy_kernel<<<nblk(S, 256), 256, 0, stream>>>(bases, Xin, S);
  zero_kernel<<<nblk(S, 256), 256, 0, stream>>>(bases + S, S);
  launch_spmv(bases + S, bases, rows, cols, vals, 1.0f, V, Cin, E, stream);
  // Bk = 2 L B(k-1) - B(k-2)
  for (int k = 2; k < K_CHEB; ++k) {
    negcopy_kernel<<<nblk(S, 256), 256, 0, stream>>>(bases + (long)k * S,
                                                     bases + (long)(k - 2) * S, S);
    launch_spmv(bases + (long)k * S, bases + (long)(k - 1) * S,
                rows, cols, vals, 2.0f, V, Cin, E, stream);
  }
  launch_gemm_fused(out, bases, S, W, bias, Rtiles, Cin, Cout, stream);
}

// STEPWISE plan: 3 rotating basis buffers, accumulate per k.
static void run_cheb_conv_step(const float* Xin, const float* W, const float* bias,
                               const int* rows, const int* cols, const float* vals,
                               float* out, float* bufA, float* bufB, float* bufC,
                               int V, int E, int Cin, int Cout, hipStream_t stream) {
  long R = (long)V * BATCH;
  int Rtiles = (int)(R >> 4);
  long outN = R * Cout;
  long xN   = R * Cin;

  bias_init_kernel<<<nblk(outN, 256), 256, 0, stream>>>(out, bias, outN, Cout - 1);
  launch_gemm_acc(out, Xin, W, Rtiles, Cin, Cout, stream);

  zero_kernel<<<nblk(xN, 256), 256, 0, stream>>>(bufA, xN);
  launch_spmv(bufA, Xin, rows, cols, vals, 1.0f, V, Cin, E, stream);
  launch_gemm_acc(out, bufA, W + (long)Cin * Cout, Rtiles, Cin, Cout, stream);

  const float* x0 = Xin;
  float* x1 = bufA;
  float* spare  = bufB;
  float* spare2 = bufC;
  for (int k = 2; k < K_CHEB; ++k) {
    float* x2 = spare;
    negcopy_kernel<<<nblk(xN, 256), 256, 0, stream>>>(x2, x0, xN);
    launch_spmv(x2, x1, rows, cols, vals, 2.0f, V, Cin, E, stream);
    launch_gemm_acc(out, x2, W + (long)k * Cin * Cout, Rtiles, Cin, Cout, stream);
    float* freed = (k == 2) ? spare2 : (float*)x0;
    spare = freed;
    x0 = x1;
    x1 = x2;
  }
}

extern "C" void kernel_launch(void* const* d_in, const int* in_sizes, int n_in,
                              void* d_out, int out_size, void* d_ws, size_t ws_size,
                              hipStream_t stream) {
  (void)in_sizes; (void)n_in; (void)out_size;

  const float* x    = (const float*)d_in[0];
  const float* w1_1 = (const float*)d_in[1];
  const float* b1_1 = (const float*)d_in[2];
  const float* w2_1 = (const float*)d_in[3];
  const float* b2_1 = (const float*)d_in[4];
  const int*   rows1 = (const int*)d_in[5];
  const int*   cols1 = (const int*)d_in[6];
  const float* vals1 = (const float*)d_in[7];
  const float* w1_2 = (const float*)d_in[8];
  const float* b1_2 = (const float*)d_in[9];
  const float* w2_2 = (const float*)d_in[10];
  const float* b2_2 = (const float*)d_in[11];
  const int*   rows2 = (const int*)d_in[12];
  const int*   cols2 = (const int*)d_in[13];
  const float* vals2 = (const float*)d_in[14];
  const float* w1_3 = (const float*)d_in[15];
  const float* b1_3 = (const float*)d_in[16];
  const float* w2_3 = (const float*)d_in[17];
  const float* b2_3 = (const float*)d_in[18];
  const int*   rows3 = (const int*)d_in[19];
  const int*   cols3 = (const int*)d_in[20];
  const float* vals3 = (const float*)d_in[21];

  const int V1 = 49152, V2 = 12288, V3 = 3072;
  const int E1 = 8 * V1, E2 = 8 * V2, E3 = 8 * V3;

  // output slices (flat, in return order)
  float* outp  = (float*)d_out;
  float* skip1 = outp;                                // [8, 49152, 32]
  float* skip2 = outp + (long)BATCH * V1 * 32;        // [8, 12288, 64]
  float* skip3 = skip2 + (long)BATCH * V2 * 64;       // [8, 3072, 128]
  float* xfin  = skip3 + (long)BATCH * V3 * 128;      // [8, 768, 128]

  const long CHEB = (long)BATCH * V1 * 32;            // 12,582,912 floats (max R*Cin)
  const long POOLSZ = (long)BATCH * V2 * 32;          // 3,145,728 floats (max pooled)

  // FUSED plan ws: K*CHEB bases + t1 + pooled
  const unsigned long long fusedBytes =
      (unsigned long long)(K_CHEB * CHEB + CHEB + POOLSZ) * 4ull;   // ~566 MB
  bool fused = ws_size >= fusedBytes;

  float* ws = (float*)d_ws;

  if (fused) {
    float* bases  = ws;
    float* t1     = bases + (long)K_CHEB * CHEB;
    float* pooled = t1 + CHEB;

    run_cheb_conv_fused(x, w1_1, b1_1, rows1, cols1, vals1, t1, bases, V1, E1, 1, 32, stream);
    elu_kernel<<<nblk((long)BATCH * V1 * 32, 256), 256, 0, stream>>>(t1, (long)BATCH * V1 * 32);
    run_cheb_conv_fused(t1, w2_1, b2_1, rows1, cols1, vals1, skip1, bases, V1, E1, 32, 32, stream);
    elu_kernel<<<nblk((long)BATCH * V1 * 32, 256), 256, 0, stream>>>(skip1, (long)BATCH * V1 * 32);
    pool4_kernel<32><<<nblk((long)BATCH * V2 * 32, 256), 256, 0, stream>>>(pooled, skip1, V2);

    run_cheb_conv_fused(pooled, w1_2, b1_2, rows2, cols2, vals2, t1, bases, V2, E2, 32, 64, stream);
    elu_kernel<<<nblk((long)BATCH * V2 * 64, 256), 256, 0, stream>>>(t1, (long)BATCH * V2 * 64);
    run_cheb_conv_fused(t1, w2_2, b2_2, rows2, cols2, vals2, skip2, bases, V2, E2, 64, 64, stream);
    elu_kernel<<<nblk((long)BATCH * V2 * 64, 256), 256, 0, stream>>>(skip2, (long)BATCH * V2 * 64);
    pool4_kernel<64><<<nblk((long)BATCH * V3 * 64, 256), 256, 0, stream>>>(pooled, skip2, V3);

    run_cheb_conv_fused(pooled, w1_3, b1_3, rows3, cols3, vals3, t1, bases, V3, E3, 64, 128, stream);
    elu_kernel<<<nblk((long)BATCH * V3 * 128, 256), 256, 0, stream>>>(t1, (long)BATCH * V3 * 128);
    run_cheb_conv_fused(t1, w2_3, b2_3, rows3, cols3, vals3, skip3, bases, V3, E3, 128, 128, stream);
    elu_kernel<<<nblk((long)BATCH * V3 * 128, 256), 256, 0, stream>>>(skip3, (long)BATCH * V3 * 128);
    pool4_kernel<128><<<nblk((long)BATCH * (V3 / 4) * 128, 256), 256, 0, stream>>>(xfin, skip3, V3 / 4);
  } else {
    float* cA = ws;
    float* cB = cA + CHEB;
    float* cC = cB + CHEB;
    float* t1 = cC + CHEB;
    float* pooled = t1 + CHEB;

    run_cheb_conv_step(x, w1_1, b1_1, rows1, cols1, vals1, t1, cA, cB, cC, V1, E1, 1, 32, stream);
    elu_kernel<<<nblk((long)BATCH * V1 * 32, 256), 256, 0, stream>>>(t1, (long)BATCH * V1 * 32);
    run_cheb_conv_step(t1, w2_1, b2_1, rows1, cols1, vals1, skip1, cA, cB, cC, V1, E1, 32, 32, stream);
    elu_kernel<<<nblk((long)BATCH * V1 * 32, 256), 256, 0, stream>>>(skip1, (long)BATCH * V1 * 32);
    pool4_kernel<32><<<nblk((long)BATCH * V2 * 32, 256), 256, 0, stream>>>(pooled, skip1, V2);

    run_cheb_conv_step(pooled, w1_2, b1_2, rows2, cols2, vals2, t1, cA, cB, cC, V2, E2, 32, 64, stream);
    elu_kernel<<<nblk((long)BATCH * V2 * 64, 256), 256, 0, stream>>>(t1, (long)BATCH * V2 * 64);
    run_cheb_conv_step(t1, w2_2, b2_2, rows2, cols2, vals2, skip2, cA, cB, cC, V2, E2, 64, 64, stream);
    elu_kernel<<<nblk((long)BATCH * V2 * 64, 256), 256, 0, stream>>>(skip2, (long)BATCH * V2 * 64);
    pool4_kernel<64><<<nblk((long)BATCH * V3 * 64, 256), 256, 0, stream>>>(pooled, skip2, V3);

    run_cheb_conv_step(pooled, w1_3, b1_3, rows3, cols3, vals3, t1, cA, cB, cC, V3, E3, 64, 128, stream);
    elu_kernel<<<nblk((long)BATCH * V3 * 128, 256), 256, 0, stream>>>(t1, (long)BATCH * V3 * 128);
    run_cheb_conv_step(t1, w2_3, b2_3, rows3, cols3, vals3, skip3, cA, cB, cC, V3, E3, 128, 128, stream);
    elu_kernel<<<nblk((long)BATCH * V3 * 128, 256), 256, 0, stream>>>(skip3, (long)BATCH * V3 * 128);
    pool4_kernel<128><<<nblk((long)BATCH * (V3 / 4) * 128, 256), 256, 0, stream>>>(xfin, skip3, V3 / 4);
  }
}